// ParallelDroplessMLP_2302102471532
// MI455X (gfx1250) — compile-verified
//
#include <hip/hip_runtime.h>
#include <hip/hip_bf16.h>
#include <cstdint>

// ---------------- problem constants ----------------
#define NUM_EXPERTS 8
#define TOP_K       2
#define HIDDEN      2048
#define FFN         8192
#define NTOK        8192              // sl*bs = 2048*4
#define NFLAT       (NTOK * TOP_K)   // 16384 expanded assignments

// Padded expanded-row capacity: each expert segment padded to 128, plus the
// dense "shared MLP" segment (expert id 8) of NTOK rows (already 128-aligned).
#define NP_MAX      25600
#define MT_MAX      (NP_MAX / 128)   // 200 M-tiles

// ---------------- GEMM tiling ----------------
#define BM 128
#define BN 128
#define BK 32
#define AP 36     // LDS pitch for A tile  [BM][AP]  (m-major, k contiguous)
#define BP 132    // LDS pitch for B tile  [BK][BP]  (k-major, n contiguous)

typedef float v2f __attribute__((ext_vector_type(2)));
typedef float v8f __attribute__((ext_vector_type(8)));

// async global->LDS copy of 16 bytes per lane (CDNA5, ASYNCcnt-tracked)
__device__ __forceinline__ void async_b128(uint32_t lds_addr, const void* gptr) {
    asm volatile("global_load_async_to_lds_b128 %0, %1, off"
                 :: "v"(lds_addr), "v"((uint64_t)(uintptr_t)gptr)
                 : "memory");
}
__device__ __forceinline__ void wait_async0() {
    asm volatile("s_wait_asynccnt 0x0" ::: "memory");
}

// =====================================================================
// Kernel 1: routing — stable counting sort by expert with 128-padding,
// plus virtual shared-MLP expert appended at the end.
// =====================================================================
__global__ __launch_bounds__(256) void route_kernel(
    const int*   __restrict__ eidx,     // [NFLAT] expert ids
    const float* __restrict__ ew,       // [NFLAT] combine weights
    int*   __restrict__ row_map,        // [NP_MAX] source token row, -1 = pad
    float* __restrict__ wgt,            // [NP_MAX] combine weight (0 for pad)
    int*   __restrict__ tile_ex)        // [MT_MAX] expert per M-tile (-1 unused, 8 shared)
{
    __shared__ int cnt[NUM_EXPERTS][256];
    __shared__ int poff[NUM_EXPERTS + 1];

    const int t = threadIdx.x;

    for (int p = t; p < NP_MAX; p += 256) { row_map[p] = -1; wgt[p] = 0.0f; }

    int local[NUM_EXPERTS];
#pragma unroll
    for (int e = 0; e < NUM_EXPERTS; ++e) local[e] = 0;
    const int base = t * (NFLAT / 256);
#pragma unroll 4
    for (int i = 0; i < (NFLAT / 256); ++i) {
        int e = eidx[base + i] & (NUM_EXPERTS - 1);
        local[e]++;
    }
#pragma unroll
    for (int e = 0; e < NUM_EXPERTS; ++e) cnt[e][t] = local[e];
    __syncthreads();

    if (t == 0) {
        int po = 0;
        for (int e = 0; e < NUM_EXPERTS; ++e) {
            poff[e] = po;
            int tot = 0;
            for (int j = 0; j < 256; ++j) { int c = cnt[e][j]; cnt[e][j] = tot; tot += c; }
            po += ((tot + 127) >> 7) << 7;   // pad segment to multiple of 128
        }
        poff[NUM_EXPERTS] = po;
    }
    __syncthreads();

#pragma unroll
    for (int e = 0; e < NUM_EXPERTS; ++e) local[e] = poff[e] + cnt[e][t];
#pragma unroll 4
    for (int i = 0; i < (NFLAT / 256); ++i) {
        int gi = base + i;
        int e  = eidx[gi] & (NUM_EXPERTS - 1);
        int d  = local[e]++;
        row_map[d] = gi >> 1;        // order // TOP_K
        wgt[d]     = ew[gi];
    }

    const int sb = poff[NUM_EXPERTS];
    for (int j = t; j < NTOK; j += 256) { row_map[sb + j] = j; wgt[sb + j] = 1.0f; }

    for (int m = t; m < MT_MAX; m += 256) {
        int p0 = m * 128;
        int e  = -1;
        if (p0 < sb) {
#pragma unroll
            for (int q = 0; q < NUM_EXPERTS; ++q)
                if (p0 >= poff[q] && p0 < poff[q + 1]) e = q;
        } else if (p0 < sb + NTOK) {
            e = NUM_EXPERTS;   // shared
        }
        tile_ex[m] = e;
    }
}

// =====================================================================
// Kernel 2: zero the output (scatter-combine accumulates into it)
// =====================================================================
__global__ __launch_bounds__(256) void zero_kernel(float* __restrict__ out)
{
    int i = (blockIdx.x * 256 + threadIdx.x) * 4;
    *(float4*)(out + i) = make_float4(0.f, 0.f, 0.f, 0.f);
}

// =====================================================================
// fp32 WMMA fragment layout (V_WMMA_F32_16X16X4_F32)
//   A 16x4 : lanes 0-15 -> {K0,K1}, lanes 16-31 -> {K2,K3}, M = lane&15
//   B 4x16 : mirrored (K split across halves/VGPRs, N = lane&15)
//   C 16x16: vgpr r -> M = r + 8*(lane>=16), N = lane&15
// =====================================================================

// =====================================================================
// Kernel 3: GEMM1  h[p,:] = gelu( gather(x)[p,:] @ W1_e )     K=HIDDEN
// Async double-buffered: global_load_async_to_lds_b128 fills LDS buffer
// (k+1) while WMMAs consume buffer (k). One barrier per chunk.
// =====================================================================
__global__ __launch_bounds__(256) void gemm1_kernel(
    const float* __restrict__ x,        // [NTOK, HIDDEN]
    const float* __restrict__ w1,       // [E, HIDDEN, FFN]
    const float* __restrict__ sw1,      // [HIDDEN, FFN]
    const int*   __restrict__ row_map,
    const int*   __restrict__ tile_ex,
    float*       __restrict__ h)        // [NP_MAX, FFN]
{
    const int mt = blockIdx.y;
    const int ex = tile_ex[mt];
    if (ex < 0) return;
    const float* __restrict__ Bmat =
        (ex == NUM_EXPERTS) ? sw1 : (w1 + (size_t)ex * ((size_t)HIDDEN * FFN));
    const int n_base = blockIdx.x * BN;

    __shared__ float As[2][BM][AP];
    __shared__ float Bs[2][BK][BP];
    __shared__ int   rows[BM];

    const int tid = threadIdx.x;
    for (int r = tid; r < BM; r += 256) rows[r] = row_map[mt * BM + r];

    const int wave = tid >> 5, lane = tid & 31;
    const int l15 = lane & 15, hi = lane >> 4;
    const int wm = (wave >> 2) * 64;    // 2 waves in M
    const int wn = (wave & 3) * 32;     // 4 waves in N

    // per-thread staging slots (4 of 1024 b128 slots for A and for B)
    int ar[4], kq[4], br[4], nq[4];
#pragma unroll
    for (int s4 = 0; s4 < 4; ++s4) {
        int s = tid + s4 * 256;
        ar[s4] = s >> 3;  kq[s4] = (s & 7)  * 4;   // A: 8 b128 per 32-f row
        br[s4] = s >> 5;  nq[s4] = (s & 31) * 4;   // B: 32 b128 per 128-f row
    }

    v8f c[4][2];
#pragma unroll
    for (int i = 0; i < 4; ++i)
#pragma unroll
        for (int j = 0; j < 2; ++j) c[i][j] = (v8f){0,0,0,0,0,0,0,0};

    __syncthreads();   // rows[] visible

    // pre-zero static pad rows (src<0) in both A buffers; async issue is
    // predicated on src>=0 so these rows stay zero for the whole kernel.
    for (int idx = tid; idx < BM * 8; idx += 256) {
        int r = idx >> 3, q = (idx & 7) * 4;
        if (rows[r] < 0) {
            *(float4*)&As[0][r][q] = make_float4(0.f, 0.f, 0.f, 0.f);
            *(float4*)&As[1][r][q] = make_float4(0.f, 0.f, 0.f, 0.f);
        }
    }
    __syncthreads();

    auto issue_async = [&](int k0, int buf) {
#pragma unroll
        for (int s4 = 0; s4 < 4; ++s4) {
            int src = rows[ar[s4]];
            if (src >= 0)
                async_b128((uint32_t)(uintptr_t)&As[buf][ar[s4]][kq[s4]],
                           x + (size_t)src * HIDDEN + k0 + kq[s4]);
            async_b128((uint32_t)(uintptr_t)&Bs[buf][br[s4]][nq[s4]],
                       Bmat + (size_t)(k0 + br[s4]) * FFN + n_base + nq[s4]);
        }
    };

    issue_async(0, 0);

    for (int k0 = 0; k0 < HIDDEN; k0 += BK) {
        const int buf = (k0 / BK) & 1;
        wait_async0();       // our buffer's DMA done (per-wave)
        __syncthreads();     // every wave's DMA done; prev compute finished
        if (k0 + BK < HIDDEN) issue_async(k0 + BK, buf ^ 1);   // overlaps WMMAs

#pragma unroll
        for (int ks = 0; ks < BK; ks += 4) {
            const int kb = ks + hi * 2;
            v2f a[4], b[2];
#pragma unroll
            for (int i = 0; i < 4; ++i)
                a[i] = *(const v2f*)&As[buf][wm + i * 16 + l15][kb];
#pragma unroll
            for (int j = 0; j < 2; ++j) {
                b[j][0] = Bs[buf][kb][wn + j * 16 + l15];
                b[j][1] = Bs[buf][kb + 1][wn + j * 16 + l15];
            }
#pragma unroll
            for (int i = 0; i < 4; ++i)
#pragma unroll
                for (int j = 0; j < 2; ++j)
                    c[i][j] = __builtin_amdgcn_wmma_f32_16x16x4_f32(
                        false, a[i], false, b[j], (short)0, c[i][j], false, false);
        }
    }

    // epilogue: exact gelu, store h
    const size_t p_base = (size_t)mt * BM;
#pragma unroll
    for (int i = 0; i < 4; ++i)
#pragma unroll
        for (int j = 0; j < 2; ++j)
#pragma unroll
            for (int r = 0; r < 8; ++r) {
                int m = wm + i * 16 + hi * 8 + r;
                int n = n_base + wn + j * 16 + l15;
                float v = c[i][j][r];
                float g = 0.5f * v * (1.0f + erff(v * 0.70710678118654752f));
                h[(p_base + m) * (size_t)FFN + n] = g;
            }
}

// =====================================================================
// Kernel 4: GEMM2  out[row_map[p],:] += wgt[p] * ( h[p,:] @ W2_e )   K=FFN
// =====================================================================
__global__ __launch_bounds__(256) void gemm2_kernel(
    const float* __restrict__ h,        // [NP_MAX, FFN]
    const float* __restrict__ w2,       // [E, FFN, HIDDEN]
    const float* __restrict__ sw2,      // [FFN, HIDDEN]
    const int*   __restrict__ row_map,
    const float* __restrict__ wgt,
    const int*   __restrict__ tile_ex,
    float*       __restrict__ out)      // [NTOK, HIDDEN]
{
    const int mt = blockIdx.y;
    const int ex = tile_ex[mt];
    if (ex < 0) return;
    const float* __restrict__ Bmat =
        (ex == NUM_EXPERTS) ? sw2 : (w2 + (size_t)ex * ((size_t)FFN * HIDDEN));
    const int n_base = blockIdx.x * BN;

    __shared__ float As[2][BM][AP];
    __shared__ float Bs[2][BK][BP];

    const int tid = threadIdx.x;
    const int wave = tid >> 5, lane = tid & 31;
    const int l15 = lane & 15, hi = lane >> 4;
    const int wm = (wave >> 2) * 64;
    const int wn = (wave & 3) * 32;
    const size_t p_base = (size_t)mt * BM;

    int ar[4], kq[4], br[4], nq[4];
#pragma unroll
    for (int s4 = 0; s4 < 4; ++s4) {
        int s = tid + s4 * 256;
        ar[s4] = s >> 3;  kq[s4] = (s & 7)  * 4;
        br[s4] = s >> 5;  nq[s4] = (s & 31) * 4;
    }

    v8f c[4][2];
#pragma unroll
    for (int i = 0; i < 4; ++i)
#pragma unroll
        for (int j = 0; j < 2; ++j) c[i][j] = (v8f){0,0,0,0,0,0,0,0};

    auto issue_async = [&](int k0, int buf) {
#pragma unroll
        for (int s4 = 0; s4 < 4; ++s4) {
            async_b128((uint32_t)(uintptr_t)&As[buf][ar[s4]][kq[s4]],
                       h + (p_base + ar[s4]) * (size_t)FFN + k0 + kq[s4]);
            async_b128((uint32_t)(uintptr_t)&Bs[buf][br[s4]][nq[s4]],
                       Bmat + (size_t)(k0 + br[s4]) * HIDDEN + n_base + nq[s4]);
        }
    };

    issue_async(0, 0);

    for (int k0 = 0; k0 < FFN; k0 += BK) {
        const int buf = (k0 / BK) & 1;
        wait_async0();
        __syncthreads();
        if (k0 + BK < FFN) issue_async(k0 + BK, buf ^ 1);   // overlaps WMMAs

#pragma unroll
        for (int ks = 0; ks < BK; ks += 4) {
            const int kb = ks + hi * 2;
            v2f a[4], b[2];
#pragma unroll
            for (int i = 0; i < 4; ++i)
                a[i] = *(const v2f*)&As[buf][wm + i * 16 + l15][kb];
#pragma unroll
            for (int j = 0; j < 2; ++j) {
                b[j][0] = Bs[buf][kb][wn + j * 16 + l15];
                b[j][1] = Bs[buf][kb + 1][wn + j * 16 + l15];
            }
#pragma unroll
            for (int i = 0; i < 4; ++i)
#pragma unroll
                for (int j = 0; j < 2; ++j)
                    c[i][j] = __builtin_amdgcn_wmma_f32_16x16x4_f32(
                        false, a[i], false, b[j], (short)0, c[i][j], false, false);
        }
    }

    // epilogue: weighted scatter-add into output
#pragma unroll
    for (int i = 0; i < 4; ++i)
#pragma unroll
        for (int j = 0; j < 2; ++j) {
            int m = wm + i * 16 + hi * 8;
            int p = (int)p_base + m;
#pragma unroll
            for (int r = 0; r < 8; ++r) {
                int pr  = p + r;
                int dst = row_map[pr];
                if (dst >= 0) {
                    int n = n_base + wn + j * 16 + l15;
                    atomicAdd(out + (size_t)dst * HIDDEN + n, wgt[pr] * c[i][j][r]);
                }
            }
        }
}

// =====================================================================
// host-side launch
// =====================================================================
extern "C" void kernel_launch(void* const* d_in, const int* in_sizes, int n_in,
                              void* d_out, int out_size, void* d_ws, size_t ws_size,
                              hipStream_t stream) {
    const float* x    = (const float*)d_in[0];   // [2048,4,2048]
    const float* ew   = (const float*)d_in[1];   // [8192,2]
    const int*   eidx = (const int*)  d_in[2];   // [8192,2]
    const float* w1   = (const float*)d_in[3];   // [8,2048,8192]
    const float* w2   = (const float*)d_in[4];   // [8,8192,2048]
    const float* sw1  = (const float*)d_in[5];   // [2048,8192]
    const float* sw2  = (const float*)d_in[6];   // [8192,2048]
    float* out = (float*)d_out;                  // [2048,4,2048]

    uint8_t* ws = (uint8_t*)d_ws;
    int*   row_map = (int*)  (ws);
    float* wgt     = (float*)(ws + (size_t)NP_MAX * 4);
    int*   tile_ex = (int*)  (ws + (size_t)NP_MAX * 8);
    float* h       = (float*)(ws + (size_t)NP_MAX * 8 + 1024); // 16B aligned

    route_kernel<<<1, 256, 0, stream>>>(eidx, ew, row_map, wgt, tile_ex);
    zero_kernel<<<(NTOK * HIDDEN) / 1024, 256, 0, stream>>>(out);
    gemm1_kernel<<<dim3(FFN / BN, MT_MAX), 256, 0, stream>>>(
        x, w1, sw1, row_map, tile_ex, h);
    gemm2_kernel<<<dim3(HIDDEN / BN, MT_MAX), 256, 0, stream>>>(
        h, w2, sw2, row_map, wgt, tile_ex, out);
}